// GAT_17343077941479
// MI455X (gfx1250) — compile-verified
//
#include <hip/hip_runtime.h>

typedef __attribute__((ext_vector_type(2))) float v2f;
typedef __attribute__((ext_vector_type(8))) float v8f;

#define NEGV (-1.0e9f)
#define SW 68   // padded LDS row stride (floats) to break bank conflicts

static __device__ __forceinline__ v8f wmma_f32_4(v2f a, v2f b, v8f c) {
    // D = A(16x4 f32) * B(4x16 f32) + C(16x16 f32)
    return __builtin_amdgcn_wmma_f32_16x16x4_f32(false, a, false, b, (short)0, c, false, false);
}

// ---- CDNA5 async global->LDS DMA helpers (ASYNCcnt tracked) -----------------
static __device__ __forceinline__ unsigned lds_off(const void* p) {
    // generic pointer to LDS: low 32 bits are the workgroup-relative LDS address
    return (unsigned)(unsigned long long)p;
}
static __device__ __forceinline__ void agl_b128(unsigned lds, const float* g) {
    asm volatile("global_load_async_to_lds_b128 %0, %1, off"
                 :: "v"(lds), "v"(g) : "memory");
}
static __device__ __forceinline__ void agl_b32(unsigned lds, const float* g) {
    asm volatile("global_load_async_to_lds_b32 %0, %1, off"
                 :: "v"(lds), "v"(g) : "memory");
}
static __device__ __forceinline__ void wait_async0() {
    asm volatile("s_wait_asynccnt 0x0" ::: "memory");
}

// ---------------------------------------------------------------------------
// Kernel 1: per-row k-th largest threshold via 32-bit radix binary search.
// ---------------------------------------------------------------------------
__global__ __launch_bounds__(128) void topk_thr_kernel(const float* __restrict__ adj,
                                                       float* __restrict__ thr,
                                                       int N, int k) {
    __shared__ unsigned s_keys[1024];
    __shared__ int s_cnt;
    long row = blockIdx.x;
    const float* rp = adj + row * (long)N;
    for (int i = threadIdx.x; i < N; i += blockDim.x) {
        unsigned u = __float_as_uint(rp[i]);
        s_keys[i] = (u & 0x80000000u) ? ~u : (u | 0x80000000u);
    }
    __syncthreads();
    unsigned lo = 0u;  // invariant: count(keys >= lo) >= k
    for (int bit = 31; bit >= 0; --bit) {
        unsigned T = lo | (1u << bit);
        if (threadIdx.x == 0) s_cnt = 0;
        __syncthreads();
        int c = 0;
        for (int i = threadIdx.x; i < N; i += blockDim.x) c += (s_keys[i] >= T) ? 1 : 0;
        atomicAdd(&s_cnt, c);
        __syncthreads();
        if (s_cnt >= k) lo = T;
        __syncthreads();
    }
    if (threadIdx.x == 0) {
        unsigned key = lo;
        unsigned bits = (key & 0x80000000u) ? (key ^ 0x80000000u) : ~key;
        thr[row] = __uint_as_float(bits);
    }
}

// ---------------------------------------------------------------------------
// Kernel 2: dual GEMM  F = A*W, R = A*rW  via WMMA f32 16x16x4.
// ---------------------------------------------------------------------------
__global__ __launch_bounds__(512) void gemm_dual_kernel(const float* __restrict__ A,
                                                        const float* __restrict__ Wm,
                                                        const float* __restrict__ Rm,
                                                        float* __restrict__ F,
                                                        float* __restrict__ R,
                                                        int Fin, int Fout) {
    int lane = threadIdx.x & 31;
    int ct   = threadIdx.x >> 5;
    int n    = lane & 15;
    int kh   = lane >> 4;
    long row0 = (long)blockIdx.x * 16;
    int  col0 = ct * 16;

    v8f cf = {};
    v8f cr = {};
    for (int k = 0; k < Fin; k += 4) {
        const float* ap = A + (row0 + n) * (long)Fin + k + 2 * kh;
        v2f a; a.x = ap[0]; a.y = ap[1];
        const float* wp = Wm + (long)(k + 2 * kh) * Fout + col0 + n;
        const float* rp = Rm + (long)(k + 2 * kh) * Fout + col0 + n;
        v2f bw; bw.x = wp[0]; bw.y = wp[Fout];
        v2f br; br.x = rp[0]; br.y = rp[Fout];
        cf = wmma_f32_4(a, bw, cf);
        cr = wmma_f32_4(a, br, cr);
    }
#pragma unroll
    for (int r = 0; r < 8; ++r) {
        long row = row0 + r + kh * 8;
        F[row * (long)Fout + col0 + n] = cf[r];
        R[row * (long)Fout + col0 + n] = cr[r];
    }
}

// ---------------------------------------------------------------------------
// Kernel 3: el/er per (node, head)
// ---------------------------------------------------------------------------
__global__ void elr_kernel(const float* __restrict__ f,
                           const float* __restrict__ al,
                           const float* __restrict__ ar,
                           float* __restrict__ el, float* __restrict__ er,
                           int total_nodes, int H, int D) {
    int idx = blockIdx.x * blockDim.x + threadIdx.x;
    if (idx >= total_nodes * H) return;
    int h = idx % H;
    long node = idx / H;
    const float* fp  = f + node * (long)(H * D) + h * D;
    const float* alp = al + h * D;
    const float* arp = ar + h * D;
    float sl = 0.f, sr = 0.f;
    for (int d = 0; d < D; ++d) { float v = fp[d]; sl += v * alp[d]; sr += v * arp[d]; }
    el[idx] = sl;
    er[idx] = sr;
}

static __device__ __forceinline__ float edge_weight(float a, float t, float e,
                                                    float m, float invl) {
    bool msk = (a >= t) && (a > 0.f);
    e = e > 0.f ? e : 0.2f * e;                 // leaky relu 0.2
    return msk ? __expf(e - m) * invl : 0.f;
}

// ---------------------------------------------------------------------------
// Layer-0 fused attention: block = 4 waves = 64 dst nodes, one head.
// u processed in 64-row blocks double-buffered in LDS via async DMA.
// ---------------------------------------------------------------------------
__global__ __launch_bounds__(128) void attn_kernel_l0(const float* __restrict__ adj,
                                                      const float* __restrict__ thr,
                                                      const float* __restrict__ el,
                                                      const float* __restrict__ er,
                                                      const float* __restrict__ f,
                                                      const float* __restrict__ res,
                                                      const float* __restrict__ bias,
                                                      float* __restrict__ out,
                                                      int N, int H, int D) {
    __shared__ __align__(16) float s_adj[2][64 * SW];
    __shared__ __align__(16) float s_fb [2][64 * SW];
    __shared__ __align__(16) float s_el [2][64];
    __shared__ __align__(16) float s_th [2][64];

    const int HD = H * D;
    int tid  = threadIdx.x;
    int lane = tid & 31;
    int w    = tid >> 5;
    int vl   = lane & 15;
    int uh   = lane >> 4;
    int nvg  = N / 64;
    int vg   = blockIdx.x % nvg;
    int h    = (blockIdx.x / nvg) % H;
    int b    = blockIdx.x / (nvg * H);
    int v0   = vg * 64;
    int vcol = w * 16 + vl;
    long base = (long)b * N;

    float er_v = er[(base + v0 + vcol) * H + h];

    auto stage1 = [&](int u0, int bs) {
#pragma unroll
        for (int j = 0; j < 8; ++j) {
            int c = tid + 128 * j;
            int row = c >> 4, c16 = c & 15;
            agl_b128(lds_off(&s_adj[bs][row * SW + c16 * 4]),
                     adj + (base + u0 + row) * (long)N + v0 + c16 * 4);
        }
        if (tid < 64) agl_b32(lds_off(&s_el[bs][tid]), el + (base + u0 + tid) * H + h);
        else          agl_b32(lds_off(&s_th[bs][tid - 64]), thr + base + u0 + (tid - 64));
    };
    auto stage2 = [&](int u0, int bs) {
        stage1(u0, bs);
#pragma unroll
        for (int j = 0; j < 8; ++j) {
            int c = tid + 128 * j;
            int row = c >> 4, c16 = c & 15;
            agl_b128(lds_off(&s_fb[bs][row * SW + c16 * 4]),
                     f + (base + u0 + row) * (long)HD + h * D + c16 * 4);
        }
    };

    // ---- pass 1: masked online max/sum over sources ----
    float m = NEGV, l = 0.f;
    stage1(0, 0);
    wait_async0();
    __syncthreads();
    int buf = 0;
    for (int ib = 0; ib < 16; ++ib) {
        if (ib + 1 < 16) stage1((ib + 1) * 64, buf ^ 1);
        for (int i = uh; i < 64; i += 2) {
            float a = s_adj[buf][i * SW + vcol];
            float t = s_th[buf][i];
            bool msk = (a >= t) && (a > 0.f);
            float e = s_el[buf][i] + er_v;
            e = e > 0.f ? e : 0.2f * e;
            float logit = msk ? e : NEGV;
            float nm = fmaxf(m, logit);
            l = l * __expf(m - nm) + (msk ? __expf(e - nm) : 0.f);
            m = nm;
        }
        wait_async0();
        __syncthreads();
        buf ^= 1;
    }
    {
        float m2 = __shfl_xor(m, 16);
        float l2 = __shfl_xor(l, 16);
        float nm = fmaxf(m, m2);
        l = l * __expf(m - nm) + l2 * __expf(m2 - nm);
        m = nm;
    }
    float invl = 1.f / fmaxf(l, 1e-9f);

    // ---- pass 2: P @ f via WMMA, K-step = 4 sources ----
    v8f acc0 = {}, acc1 = {}, acc2 = {}, acc3 = {};
    stage2(0, 0);
    wait_async0();
    __syncthreads();
    buf = 0;
    for (int ib = 0; ib < 16; ++ib) {
        if (ib + 1 < 16) stage2((ib + 1) * 64, buf ^ 1);
#pragma unroll
        for (int ks = 0; ks < 16; ++ks) {
            int ul = ks * 4 + 2 * uh;
            float a0 = s_adj[buf][ul * SW + vcol];
            float a1 = s_adj[buf][(ul + 1) * SW + vcol];
            v2f Aop;
            Aop.x = edge_weight(a0, s_th[buf][ul],     s_el[buf][ul]     + er_v, m, invl);
            Aop.y = edge_weight(a1, s_th[buf][ul + 1], s_el[buf][ul + 1] + er_v, m, invl);
            const float* fr0 = &s_fb[buf][ul * SW + vl];
            const float* fr1 = fr0 + SW;
            v2f B0; B0.x = fr0[0];  B0.y = fr1[0];
            v2f B1; B1.x = fr0[16]; B1.y = fr1[16];
            v2f B2; B2.x = fr0[32]; B2.y = fr1[32];
            v2f B3; B3.x = fr0[48]; B3.y = fr1[48];
            acc0 = wmma_f32_4(Aop, B0, acc0);
            acc1 = wmma_f32_4(Aop, B1, acc1);
            acc2 = wmma_f32_4(Aop, B2, acc2);
            acc3 = wmma_f32_4(Aop, B3, acc3);
        }
        wait_async0();
        __syncthreads();
        buf ^= 1;
    }

    // ---- epilogue: + residual + bias, ELU ----
#pragma unroll
    for (int r = 0; r < 8; ++r) {
        int v = v0 + w * 16 + r + uh * 8;
        long o = (base + v) * (long)HD + h * D;
        float vals[4] = {acc0[r], acc1[r], acc2[r], acc3[r]};
#pragma unroll
        for (int t = 0; t < 4; ++t) {
            int d = 16 * t + vl;
            float x = vals[t] + res[o + d] + bias[h * D + d];
            out[o + d] = x > 0.f ? x : __expf(x) - 1.f;
        }
    }
}

// ---------------------------------------------------------------------------
// Layer-1 fused attention: all 4 heads; pass-1 does all heads in one sweep
// (el staged as b128 of 4 head values per node); head-mean in epilogue.
// ---------------------------------------------------------------------------
__global__ __launch_bounds__(128) void attn_kernel_l1(const float* __restrict__ adj,
                                                      const float* __restrict__ thr,
                                                      const float* __restrict__ el,
                                                      const float* __restrict__ er,
                                                      const float* __restrict__ f,
                                                      const float* __restrict__ res,
                                                      const float* __restrict__ bias,
                                                      float* __restrict__ out,
                                                      int N, int H, int D) {
    __shared__ __align__(16) float s_adj[2][64 * SW];
    __shared__ __align__(16) float s_fb [2][64 * SW];
    __shared__ __align__(16) float s_el4[2][64 * 4];
    __shared__ __align__(16) float s_th [2][64];

    const int HD = H * D;
    int tid  = threadIdx.x;
    int lane = tid & 31;
    int w    = tid >> 5;
    int vl   = lane & 15;
    int uh   = lane >> 4;
    int nvg  = N / 64;
    int vg   = blockIdx.x % nvg;
    int b    = blockIdx.x / nvg;
    int v0   = vg * 64;
    int vcol = w * 16 + vl;
    long base = (long)b * N;

    float erv[4], m[4], lsum[4], invl[4];
#pragma unroll
    for (int hh = 0; hh < 4; ++hh) {
        erv[hh] = er[(base + v0 + vcol) * H + hh];
        m[hh] = NEGV;
        lsum[hh] = 0.f;
    }

    auto stage1 = [&](int u0, int bs) {
#pragma unroll
        for (int j = 0; j < 8; ++j) {
            int c = tid + 128 * j;
            int row = c >> 4, c16 = c & 15;
            agl_b128(lds_off(&s_adj[bs][row * SW + c16 * 4]),
                     adj + (base + u0 + row) * (long)N + v0 + c16 * 4);
        }
        if (tid < 64) agl_b128(lds_off(&s_el4[bs][tid * 4]), el + (base + u0 + tid) * H);
        else          agl_b32(lds_off(&s_th[bs][tid - 64]), thr + base + u0 + (tid - 64));
    };
    auto stage2 = [&](int u0, int bs, int h) {
        stage1(u0, bs);
#pragma unroll
        for (int j = 0; j < 8; ++j) {
            int c = tid + 128 * j;
            int row = c >> 4, c16 = c & 15;
            agl_b128(lds_off(&s_fb[bs][row * SW + c16 * 4]),
                     f + (base + u0 + row) * (long)HD + h * D + c16 * 4);
        }
    };

    // ---- pass 1 (all heads at once) ----
    stage1(0, 0);
    wait_async0();
    __syncthreads();
    int buf = 0;
    for (int ib = 0; ib < 16; ++ib) {
        if (ib + 1 < 16) stage1((ib + 1) * 64, buf ^ 1);
        for (int i = uh; i < 64; i += 2) {
            float a = s_adj[buf][i * SW + vcol];
            float t = s_th[buf][i];
            bool msk = (a >= t) && (a > 0.f);
#pragma unroll
            for (int hh = 0; hh < 4; ++hh) {
                float e = s_el4[buf][i * 4 + hh] + erv[hh];
                e = e > 0.f ? e : 0.2f * e;
                float logit = msk ? e : NEGV;
                float nm = fmaxf(m[hh], logit);
                lsum[hh] = lsum[hh] * __expf(m[hh] - nm) + (msk ? __expf(e - nm) : 0.f);
                m[hh] = nm;
            }
        }
        wait_async0();
        __syncthreads();
        buf ^= 1;
    }
#pragma unroll
    for (int hh = 0; hh < 4; ++hh) {
        float m2 = __shfl_xor(m[hh], 16);
        float l2 = __shfl_xor(lsum[hh], 16);
        float nm = fmaxf(m[hh], m2);
        lsum[hh] = lsum[hh] * __expf(m[hh] - nm) + l2 * __expf(m2 - nm);
        m[hh] = nm;
        invl[hh] = 1.f / fmaxf(lsum[hh], 1e-9f);
    }

    // ---- pass 2: per head, accumulate normalized aggregation ----
    v8f acc0 = {}, acc1 = {}, acc2 = {}, acc3 = {};
    for (int h = 0; h < 4; ++h) {
        stage2(0, 0, h);
        wait_async0();
        __syncthreads();
        buf = 0;
        for (int ib = 0; ib < 16; ++ib) {
            if (ib + 1 < 16) stage2((ib + 1) * 64, buf ^ 1, h);
#pragma unroll
            for (int ks = 0; ks < 16; ++ks) {
                int ul = ks * 4 + 2 * uh;
                float a0 = s_adj[buf][ul * SW + vcol];
                float a1 = s_adj[buf][(ul + 1) * SW + vcol];
                v2f Aop;
                Aop.x = edge_weight(a0, s_th[buf][ul],
                                    s_el4[buf][ul * 4 + h] + erv[h], m[h], invl[h]);
                Aop.y = edge_weight(a1, s_th[buf][ul + 1],
                                    s_el4[buf][(ul + 1) * 4 + h] + erv[h], m[h], invl[h]);
                const float* fr0 = &s_fb[buf][ul * SW + vl];
                const float* fr1 = fr0 + SW;
                v2f B0; B0.x = fr0[0];  B0.y = fr1[0];
                v2f B1; B1.x = fr0[16]; B1.y = fr1[16];
                v2f B2; B2.x = fr0[32]; B2.y = fr1[32];
                v2f B3; B3.x = fr0[48]; B3.y = fr1[48];
                acc0 = wmma_f32_4(Aop, B0, acc0);
                acc1 = wmma_f32_4(Aop, B1, acc1);
                acc2 = wmma_f32_4(Aop, B2, acc2);
                acc3 = wmma_f32_4(Aop, B3, acc3);
            }
            wait_async0();
            __syncthreads();
            buf ^= 1;
        }
    }

    // ---- epilogue: + sum_h(residual + bias), mean over heads ----
#pragma unroll
    for (int r = 0; r < 8; ++r) {
        int v = v0 + w * 16 + r + uh * 8;
        long ro = (base + v) * (long)HD;
        float vals[4] = {acc0[r], acc1[r], acc2[r], acc3[r]};
#pragma unroll
        for (int t = 0; t < 4; ++t) {
            int d = 16 * t + vl;
            float x = vals[t];
#pragma unroll
            for (int h = 0; h < 4; ++h)
                x += res[ro + h * D + d] + bias[h * D + d];
            out[(base + v) * (long)D + d] = 0.25f * x;
        }
    }
}

// ---------------------------------------------------------------------------
extern "C" void kernel_launch(void* const* d_in, const int* in_sizes, int n_in,
                              void* d_out, int out_size, void* d_ws, size_t ws_size,
                              hipStream_t stream) {
    (void)in_sizes; (void)n_in; (void)out_size; (void)ws_size;
    const float* seg = (const float*)d_in[0];
    const float* adj = (const float*)d_in[1];
    const float* W0  = (const float*)d_in[2];
    const float* al0 = (const float*)d_in[3];
    const float* ar0 = (const float*)d_in[4];
    const float* rW0 = (const float*)d_in[5];
    const float* b0  = (const float*)d_in[6];
    const float* W1  = (const float*)d_in[7];
    const float* al1 = (const float*)d_in[8];
    const float* ar1 = (const float*)d_in[9];
    const float* rW1 = (const float*)d_in[10];
    const float* b1  = (const float*)d_in[11];

    const int B = 16, N = 1024, H = 4, D = 64, HD = 256;
    const int K = (32 * 32) / 6;  // 170

    float* ws = (float*)d_ws;
    size_t off = 0;
    float* thr = ws + off; off += (size_t)B * N;
    float* f0  = ws + off; off += (size_t)B * N * HD;
    float* r0  = ws + off; off += (size_t)B * N * HD;
    float* el0 = ws + off; off += (size_t)B * N * H;
    float* er0 = ws + off; off += (size_t)B * N * H;
    float* fea = ws + off; off += (size_t)B * N * HD;
    float* f1  = ws + off; off += (size_t)B * N * HD;
    float* r1  = ws + off; off += (size_t)B * N * HD;
    float* el1 = ws + off; off += (size_t)B * N * H;
    float* er1 = ws + off; off += (size_t)B * N * H;

    topk_thr_kernel<<<B * N, 128, 0, stream>>>(adj, thr, N, K);
    gemm_dual_kernel<<<B * N / 16, 512, 0, stream>>>(seg, W0, rW0, f0, r0, 64, HD);
    elr_kernel<<<(B * N * H + 255) / 256, 256, 0, stream>>>(f0, al0, ar0, el0, er0, B * N, H, D);
    attn_kernel_l0<<<B * H * (N / 64), 128, 0, stream>>>(adj, thr, el0, er0, f0, r0, b0, fea, N, H, D);
    gemm_dual_kernel<<<B * N / 16, 512, 0, stream>>>(fea, W1, rW1, f1, r1, HD, HD);
    elr_kernel<<<(B * N * H + 255) / 256, 256, 0, stream>>>(f1, al1, ar1, el1, er1, B * N, H, D);
    attn_kernel_l1<<<B * (N / 64), 128, 0, stream>>>(adj, thr, el1, er1, f1, r1, b1, (float*)d_out, N, H, D);
}